// Slot_Merger_Cosine_avg_46986942218270
// MI455X (gfx1250) — compile-verified
//
#include <hip/hip_runtime.h>

typedef float v2f __attribute__((ext_vector_type(2)));
typedef float v8f __attribute__((ext_vector_type(8)));

#define NSLOT 9
#define NDIM 64
#define ROWF 576              // 9*64 floats per batch
#define PADOFF 576            // zeroed pad region in LDS (16 floats)
#define LDS_STRIDE 592        // 576 + 16 pad, keeps 16B alignment per wave
#define WAVES_PER_BLOCK 8
#define EPSF 1e-8f
#define THRESH 0.9f

__global__ __launch_bounds__(256)
void slot_merger_cosine_kernel(const float* __restrict__ slots,
                               float* __restrict__ out_slots,
                               float* __restrict__ out_mask,
                               int B, int nwaves)
{
    __shared__ float sm_all[WAVES_PER_BLOCK * LDS_STRIDE];
    const int lane = threadIdx.x & 31;
    const int wid  = threadIdx.x >> 5;
    float* sm = sm_all + wid * LDS_STRIDE;
    const int wave_gid = blockIdx.x * WAVES_PER_BLOCK + wid;

    // zero the pad region once (stays zero for the whole kernel)
    if (lane < 16) sm[PADOFF + lane] = 0.0f;

    const int m    = lane & 15;   // WMMA row (A) / col (B) index
    const int half = lane >> 4;   // which K-half this lane supplies
    const bool valid = (m < NSLOT);
    const int d0 = lane * 2;      // this lane's 2 dims of D=64

    for (int b = wave_gid; b < B; b += nwaves) {
        const float* src = slots + (size_t)b * ROWF;

        // prefetch next batch this wave will process
        {
            int bn = b + nwaves;
            if (bn < B)
                __builtin_prefetch(slots + (size_t)bn * ROWF + lane * 18, 0, 3);
        }

        // ---- stage batch into LDS: 144 x float4 coalesced ----
        #pragma unroll
        for (int t = 0; t < 5; ++t) {
            int idx = t * 32 + lane;
            if (idx < 144)
                ((float4*)sm)[idx] = ((const float4*)src)[idx];
        }
        asm volatile("s_wait_dscnt 0x0" ::: "memory");

        // ---- Gram matrix G = S * S^T (S zero-padded to 16x64) ----
        // 16 x V_WMMA_F32_16X16X4_F32, full fp32 accumulation
        v8f c = {0.f, 0.f, 0.f, 0.f, 0.f, 0.f, 0.f, 0.f};
        #pragma unroll
        for (int k0 = 0; k0 < NDIM; k0 += 4) {
            // A (16x4): lane m supplies S[m][k0+2*half .. +1] (contiguous pair)
            int aoff = valid ? (m * NDIM + k0 + half * 2) : PADOFF;
            v2f afrag = *(const v2f*)(sm + aoff);
            // B (4x16): lane n=m supplies S[n][k0+half] and S[n][k0+half+2]
            int b0 = valid ? (m * NDIM + k0 + half)     : PADOFF;
            int b1 = valid ? (m * NDIM + k0 + half + 2) : PADOFF;
            v2f bfrag; bfrag.x = sm[b0]; bfrag.y = sm[b1];
            c = __builtin_amdgcn_wmma_f32_16x16x4_f32(false, afrag, false, bfrag,
                                                      (short)0, c, false, false);
        }

        // ---- sim row for lane i (G symmetric: lane i's column == row i) ----
        // row 8 entry: G[8][i] lives in lane i+16, c[0] -> swizzle SWAPX16
        float g8 = __int_as_float(
            __builtin_amdgcn_ds_swizzle(__float_as_int(c[0]), 0x401f));

        // diagonal (uniform broadcast) -> norms
        float dsq[9];
        dsq[0] = __int_as_float(__builtin_amdgcn_readlane(__float_as_int(c[0]), 0));
        dsq[1] = __int_as_float(__builtin_amdgcn_readlane(__float_as_int(c[1]), 1));
        dsq[2] = __int_as_float(__builtin_amdgcn_readlane(__float_as_int(c[2]), 2));
        dsq[3] = __int_as_float(__builtin_amdgcn_readlane(__float_as_int(c[3]), 3));
        dsq[4] = __int_as_float(__builtin_amdgcn_readlane(__float_as_int(c[4]), 4));
        dsq[5] = __int_as_float(__builtin_amdgcn_readlane(__float_as_int(c[5]), 5));
        dsq[6] = __int_as_float(__builtin_amdgcn_readlane(__float_as_int(c[6]), 6));
        dsq[7] = __int_as_float(__builtin_amdgcn_readlane(__float_as_int(c[7]), 7));
        dsq[8] = __int_as_float(__builtin_amdgcn_readlane(__float_as_int(c[0]), 24));

        float nrm[9];
        #pragma unroll
        for (int j = 0; j < 9; ++j) nrm[j] = sqrtf(dsq[j]);

        // this lane's own norm (select from uniform diagonal)
        float mynsq = dsq[8];
        #pragma unroll
        for (int j = 0; j < 8; ++j) mynsq = (lane == j) ? dsq[j] : mynsq;
        float myn = sqrtf(mynsq);

        // merge mask row for lane i:  G[i][j] > 0.9*(n_i*n_j + eps)
        float colv[9] = { c[0], c[1], c[2], c[3], c[4], c[5], c[6], c[7], g8 };
        unsigned mask = 0u;
        #pragma unroll
        for (int j = 0; j < 9; ++j)
            mask |= (colv[j] > THRESH * (nrm[j] * myn + EPSF)) ? (1u << j) : 0u;

        int nums  = __popc(mask);              // row sum of boolean mask
        int first = __ffs(mask) - 1;           // first nonzero (diag guarantees >=0)
        unsigned rowinfo = mask | ((unsigned)nums << 12)
                                | ((unsigned)(first & 0xF) << 20);

        // ---- dest-column logic: this lane is destination j = lane ----
        unsigned killed = 0u;
        int lw = 0;  // last writer (diagonal guarantees one always exists)
        #pragma unroll
        for (int i = 0; i < 9; ++i) {
            unsigned ri = (unsigned)__builtin_amdgcn_readlane((int)rowinfo, i);
            unsigned mi = ri & 0x1FFu;
            int numi    = (int)((ri >> 12) & 0xFu);
            int fi      = (int)((ri >> 20) & 0xFu);
            unsigned bit = (mi >> lane) & 1u;
            unsigned wij = (numi > 1) ? bit : ((i == lane) ? 1u : 0u);
            killed |= ((numi > 1) && (lane != fi)) ? bit : 0u;
            lw = wij ? i : lw;
        }

        if (lane < NSLOT)
            out_mask[(size_t)b * NSLOT + lane] = killed ? 0.0f : 1.0f;

        // ---- final rows: out[j] = writer_vals[last_writer[j]] ----
        float* dst = out_slots + (size_t)b * ROWF;
        #pragma unroll
        for (int j = 0; j < NSLOT; ++j) {
            int w = __builtin_amdgcn_readlane(lw, j);                       // uniform
            unsigned rw = (unsigned)__builtin_amdgcn_readlane((int)rowinfo, w);
            unsigned mw = rw & 0x1FFu;
            int numw = (int)((rw >> 12) & 0xFu);
            v2f val;
            if (numw > 1) {   // wave-uniform branch
                float wgt = 1.0f / ((float)numw + EPSF);
                v2f acc = {0.f, 0.f};
                #pragma unroll
                for (int k = 0; k < NSLOT; ++k) {
                    float wk = ((mw >> k) & 1u) ? wgt : 0.0f;   // scalar select
                    v2f r = *(const v2f*)(sm + k * NDIM + d0);
                    acc += r * wk;
                }
                val = acc;
            } else {
                val = *(const v2f*)(sm + w * NDIM + d0);
            }
            *(v2f*)(dst + j * NDIM + d0) = val;   // 256B coalesced row store
        }
    }
}

extern "C" void kernel_launch(void* const* d_in, const int* in_sizes, int n_in,
                              void* d_out, int out_size, void* d_ws, size_t ws_size,
                              hipStream_t stream) {
    const float* slots = (const float*)d_in[0];
    int B = in_sizes[0] / (NSLOT * NDIM);
    float* out       = (float*)d_out;
    float* out_slots = out;
    float* out_mask  = out + (size_t)B * NSLOT * NDIM;

    int blocks = 2048;                       // 16384 waves; B=131072 -> 8 batches/wave
    int nwaves = blocks * WAVES_PER_BLOCK;
    slot_merger_cosine_kernel<<<blocks, 256, 0, stream>>>(slots, out_slots,
                                                          out_mask, B, nwaves);
}